// AttentionLayer_15161234555369
// MI455X (gfx1250) — compile-verified
//
#include <hip/hip_runtime.h>
#include <math.h>

#define B_N   128
#define S_N   2048
#define D     1024
#define DH    512
#define W_WIN 64
#define WL    129
#define EPSV  1e-14f

typedef float v2f __attribute__((ext_vector_type(2)));
typedef float v8f __attribute__((ext_vector_type(8)));
typedef int   v4i __attribute__((vector_size(16)));

#if __has_builtin(__builtin_amdgcn_global_load_async_to_lds_b128)
#define HAS_ASYNC_LDS 1
#else
#define HAS_ASYNC_LDS 0
#endif

#if HAS_ASYNC_LDS
__device__ __forceinline__ void async_copy16(const float* g, float* l) {
  // 16 bytes per lane, global -> LDS, tracked by ASYNCcnt
  __builtin_amdgcn_global_load_async_to_lds_b128((v4i*)g, (v4i*)l, 0, 0);
}
__device__ __forceinline__ void wait_async_le(int n) {
#if __has_builtin(__builtin_amdgcn_s_wait_asynccnt)
  if (n == 0) __builtin_amdgcn_s_wait_asynccnt(0);
  else        __builtin_amdgcn_s_wait_asynccnt(8);
#else
  if (n == 0) asm volatile("s_wait_asynccnt 0x0" ::: "memory");
  else        asm volatile("s_wait_asynccnt 0x8" ::: "memory");
#endif
  asm volatile("" ::: "memory");
}
// issue one 4KB row (1024 floats): 8 x (32 lanes x 16B)
__device__ __forceinline__ void issue_row_async(const float* grow, float* lrow, int lane) {
#pragma unroll
  for (int i = 0; i < 8; ++i)
    async_copy16(grow + lane * 4 + i * 128, lrow + lane * 4 + i * 128);
}
#endif

// ---------------- lengths[b] = sum_s !mask[s,b] ----------------
__global__ void lengths_kernel(const unsigned char* __restrict__ mask,
                               float* __restrict__ lengths) {
  const int b = blockIdx.x, t = threadIdx.x;
  int cnt = 0;
  for (int s = t; s < S_N; s += 256) cnt += (mask[s * B_N + b] == 0) ? 1 : 0;
  __shared__ int red[256];
  red[t] = cnt; __syncthreads();
  for (int off = 128; off > 0; off >>= 1) {
    if (t < off) red[t] += red[t + off];
    __syncthreads();
  }
  if (t == 0) lengths[b] = (float)red[0];
}

// ---------------- X[m,n] = sum_k A[m,k] * Bm[n,k]  (fp32 WMMA) ----------------
// block = 128 threads (4 waves); each wave owns one 16x16 output tile.
__global__ void gemm_at_bt_kernel(const float* __restrict__ A,
                                  const float* __restrict__ Bm,
                                  float* __restrict__ X,
                                  int N, int K) {
  const int lane = threadIdx.x & 31;
  const int wave = threadIdx.x >> 5;
  const int m0 = blockIdx.y * 16;
  const int n0 = (blockIdx.x * 4 + wave) * 16;
  const int kk = (lane >> 4) * 2;           // lanes 0-15: K 0,1 ; lanes 16-31: K 2,3
  const float* arow = A  + (size_t)(m0 + (lane & 15)) * K + kk;
  const float* brow = Bm + (size_t)(n0 + (lane & 15)) * K + kk;
  v8f acc = {0.f, 0.f, 0.f, 0.f, 0.f, 0.f, 0.f, 0.f};
#pragma unroll 8
  for (int k0 = 0; k0 < K; k0 += 4) {
    v2f a = *(const v2f*)(arow + k0);
    v2f b = *(const v2f*)(brow + k0);
    acc = __builtin_amdgcn_wmma_f32_16x16x4_f32(false, a, false, b,
                                                (short)0, acc, false, false);
  }
  const int col   = n0 + (lane & 15);
  const int mbase = m0 + ((lane >> 4) << 3);
#pragma unroll
  for (int v = 0; v < 8; ++v)
    X[(size_t)(mbase + v) * N + col] = acc[v];
}

// ---------------- H[m,n] = tanh( sum_k A[m,k]*Bm[n,k] + bias[n] )  (fp32 WMMA) ----------------
__global__ void gemm_bias_tanh_kernel(const float* __restrict__ A,
                                      const float* __restrict__ Bm,
                                      const float* __restrict__ bias,
                                      float* __restrict__ H,
                                      int N, int K) {
  const int lane = threadIdx.x & 31;
  const int wave = threadIdx.x >> 5;
  const int m0 = blockIdx.y * 16;
  const int n0 = (blockIdx.x * 4 + wave) * 16;
  const int kk = (lane >> 4) * 2;
  const float* arow = A  + (size_t)(m0 + (lane & 15)) * K + kk;
  const float* brow = Bm + (size_t)(n0 + (lane & 15)) * K + kk;
  v8f acc = {0.f, 0.f, 0.f, 0.f, 0.f, 0.f, 0.f, 0.f};
#pragma unroll 8
  for (int k0 = 0; k0 < K; k0 += 4) {
    v2f a = *(const v2f*)(arow + k0);
    v2f b = *(const v2f*)(brow + k0);
    acc = __builtin_amdgcn_wmma_f32_16x16x4_f32(false, a, false, b,
                                                (short)0, acc, false, false);
  }
  const int col   = n0 + (lane & 15);
  const int mbase = m0 + ((lane >> 4) << 3);
  const float bv  = bias[col];
#pragma unroll
  for (int v = 0; v < 8; ++v)
    H[(size_t)(mbase + v) * N + col] = tanhf(acc[v] + bv);
}

// ---------------- out[m,n] = tanh( Cc[m,:]·Wo[n,0:1024] + inp[m,:]·Wo[n,1024:2048] ) ----------------
__global__ void gemm_out_kernel(const float* __restrict__ Cc,
                                const float* __restrict__ inp,
                                const float* __restrict__ Wo,
                                float* __restrict__ out) {
  const int lane = threadIdx.x & 31;
  const int wave = threadIdx.x >> 5;
  const int m0 = blockIdx.y * 16;
  const int n0 = (blockIdx.x * 4 + wave) * 16;
  const int kk = (lane >> 4) * 2;
  const float* arow0 = Cc  + (size_t)(m0 + (lane & 15)) * D + kk;
  const float* arow1 = inp + (size_t)(m0 + (lane & 15)) * D + kk;
  const float* brow  = Wo  + (size_t)(n0 + (lane & 15)) * (2 * D) + kk;
  v8f acc = {0.f, 0.f, 0.f, 0.f, 0.f, 0.f, 0.f, 0.f};
#pragma unroll 8
  for (int k0 = 0; k0 < D; k0 += 4) {
    v2f a = *(const v2f*)(arow0 + k0);
    v2f b = *(const v2f*)(brow + k0);
    acc = __builtin_amdgcn_wmma_f32_16x16x4_f32(false, a, false, b,
                                                (short)0, acc, false, false);
  }
#pragma unroll 8
  for (int k0 = 0; k0 < D; k0 += 4) {
    v2f a = *(const v2f*)(arow1 + k0);
    v2f b = *(const v2f*)(brow + D + k0);
    acc = __builtin_amdgcn_wmma_f32_16x16x4_f32(false, a, false, b,
                                                (short)0, acc, false, false);
  }
  const int col   = n0 + (lane & 15);
  const int mbase = m0 + ((lane >> 4) << 3);
#pragma unroll
  for (int v = 0; v < 8; ++v)
    out[(size_t)(mbase + v) * D + col] = tanhf(acc[v]);
}

// ---------------- p, window_start from h = tanh(x W1^T + b1) ----------------
__global__ void pos_kernel(const float* __restrict__ H,     // [B, DH]
                           const float* __restrict__ W2,    // [DH]
                           const float* __restrict__ b2,    // [1]
                           const float* __restrict__ lengths,
                           float* __restrict__ pbuf,
                           int*   __restrict__ wsbuf,
                           float* __restrict__ out_ws,
                           float* __restrict__ out_we) {
  const int b = blockIdx.x, t = threadIdx.x;
  __shared__ float red[256];
  float part = 0.f;
  for (int j = t; j < DH; j += 256) part += H[(size_t)b * DH + j] * W2[j];
  red[t] = part; __syncthreads();
  for (int off = 128; off > 0; off >>= 1) {
    if (t < off) red[t] += red[t + off];
    __syncthreads();
  }
  if (t == 0) {
    const float s2   = red[0] + b2[0];
    const float psig = 1.f / (1.f + expf(-s2));
    const float L    = lengths[b];
    const float p    = (float)W_WIN + L * psig;
    const float wsf  = rintf(p - (float)W_WIN);   // round-half-even, matches jnp.round
    const int   wsi  = (int)wsf;
    pbuf[b]   = p;
    wsbuf[b]  = wsi;
    out_ws[b] = (float)wsi;
    out_we[b] = (float)(wsi + WL);
  }
}

// ---------------- scores -> softmax*gaussian -> context ----------------
__global__ void attn_kernel(const float* __restrict__ X,
                            const float* __restrict__ src,     // [S,B,D]
                            const float* __restrict__ lengths,
                            const float* __restrict__ pbuf,
                            const int*   __restrict__ wsbuf,
                            float* __restrict__ out_a,
                            float* __restrict__ cbuf) {
  const int b = blockIdx.x, t = threadIdx.x;
  const int lane = t & 31, wave = t >> 5;
  __shared__ float xs[D];
  __shared__ float sc[WL];
  __shared__ float av[WL];
  __shared__ float red[256];
#if HAS_ASYNC_LDS
  __shared__ float sbuf[8][2][D];                 // per-wave double buffer, 64KB
#endif

  const int   wsi = wsbuf[b];
  const float p   = pbuf[b];
  const float L   = lengths[b];
  const float* srow0 = src + ((size_t)wsi * B_N + b) * D;   // row stride B_N*D

#if HAS_ASYNC_LDS
  // stage x row (4KB) via async copy: 256 threads x 16B
  async_copy16(X + (size_t)b * D + t * 4, xs + t * 4);
  // prefetch this wave's first selection row
  {
    int w0 = wave;
    if (w0 < WL) issue_row_async(srow0 + (size_t)w0 * B_N * D, &sbuf[wave][0][0], lane);
  }
  wait_async_le(8);            // x-row done (1 op/lane), row prefetch may remain
  __syncthreads();
#else
  for (int k = t; k < D; k += 256) xs[k] = X[(size_t)b * D + k];
  __syncthreads();
#endif

  // wave-cooperative dot per window position
#if HAS_ASYNC_LDS
  int buf = 0;
  for (int w = wave; w < WL; w += 8) {
    const int wn = w + 8;
    if (wn < WL) issue_row_async(srow0 + (size_t)wn * B_N * D, &sbuf[wave][buf ^ 1][0], lane);
    wait_async_le(wn < WL ? 8 : 0);     // in-order: current row's 8 ops retired
    const float* lrow = &sbuf[wave][buf][0];
    float part = 0.f;
    for (int k = lane; k < D; k += 32) part += xs[k] * lrow[k];
    for (int off = 16; off > 0; off >>= 1) part += __shfl_xor(part, off, 32);
    if (lane == 0) {
      const float pos = (float)(wsi + w);
      const bool valid = (pos >= (float)W_WIN) && (pos < L + (float)W_WIN);
      sc[w] = valid ? part : EPSV;
    }
    buf ^= 1;
  }
#else
  for (int w = wave; w < WL; w += 8) {
    const float* row = srow0 + (size_t)w * B_N * D;
    float part = 0.f;
    for (int k = lane; k < D; k += 32) part += xs[k] * row[k];
    for (int off = 16; off > 0; off >>= 1) part += __shfl_xor(part, off, 32);
    if (lane == 0) {
      const float pos = (float)(wsi + w);
      const bool valid = (pos >= (float)W_WIN) && (pos < L + (float)W_WIN);
      sc[w] = valid ? part : EPSV;
    }
  }
#endif
  __syncthreads();

  // block softmax over WL=129, then * gaussian
  const float v = (t < WL) ? sc[t] : -INFINITY;
  red[t] = v; __syncthreads();
  for (int off = 128; off > 0; off >>= 1) {
    if (t < off) red[t] = fmaxf(red[t], red[t + off]);
    __syncthreads();
  }
  const float mx = red[0]; __syncthreads();
  const float e = (t < WL) ? expf(v - mx) : 0.f;
  red[t] = e; __syncthreads();
  for (int off = 128; off > 0; off >>= 1) {
    if (t < off) red[t] += red[t + off];
    __syncthreads();
  }
  const float sum = red[0];
  if (t < WL) {
    const float pos = (float)(wsi + t);
    const float g = expf(-(pos - p) * (pos - p) / 2048.0f);  // 2*STD_SQ = 2*(W/2)^2
    const float a = (e / sum) * g;
    av[t] = a;
    out_a[(size_t)b * WL + t] = a;
  }
  __syncthreads();

  // context: c[b,d] = sum_w a[w] * selection[w,d]   (rows hot in L2 after score pass)
  for (int d = t; d < D; d += 256) {
    float acc = 0.f;
    for (int w = 0; w < WL; ++w)
      acc += av[w] * srow0[(size_t)w * B_N * D + d];
    cbuf[(size_t)b * D + d] = acc;
  }
}

extern "C" void kernel_launch(void* const* d_in, const int* in_sizes, int n_in,
                              void* d_out, int out_size, void* d_ws, size_t ws_size,
                              hipStream_t stream) {
  const float*         inp  = (const float*)d_in[0];
  const float*         src  = (const float*)d_in[1];
  const unsigned char* mask = (const unsigned char*)d_in[2];
  const float*         Wi   = (const float*)d_in[3];
  const float*         Wo   = (const float*)d_in[4];
  const float*         W1   = (const float*)d_in[5];
  const float*         b1   = (const float*)d_in[6];
  const float*         W2   = (const float*)d_in[7];
  const float*         b2   = (const float*)d_in[8];

  float* out    = (float*)d_out;
  float* out_o  = out;                       // [128*1024]
  float* out_a  = out + 131072;              // [128*129]
  float* out_ws = out + 147584;              // [128]
  float* out_we = out + 147712;              // [128]

  float* ws      = (float*)d_ws;
  float* xbuf    = ws;                       // 131072 floats
  float* cbuf    = ws + 131072;              // 131072 floats
  float* hbuf    = ws + 262144;              // 65536 floats  [B, DH]
  float* lengths = ws + 327680;              // 128
  float* pbuf    = ws + 327808;              // 128
  int*   wsbuf   = (int*)(ws + 327936);      // 128

  lengths_kernel<<<B_N, 256, 0, stream>>>(mask, lengths);

  // x = input @ Wi^T   (M=128, N=1024, K=1024)
  dim3 g1(D / 64, B_N / 16);
  gemm_at_bt_kernel<<<g1, 128, 0, stream>>>(inp, Wi, xbuf, D, D);

  // h = tanh(x @ W1^T + b1)   (M=128, N=512, K=1024)
  dim3 g2(DH / 64, B_N / 16);
  gemm_bias_tanh_kernel<<<g2, 128, 0, stream>>>(xbuf, W1, b1, hbuf, DH, D);

  pos_kernel<<<B_N, 256, 0, stream>>>(hbuf, W2, b2, lengths,
                                      pbuf, wsbuf, out_ws, out_we);

  attn_kernel<<<B_N, 256, 0, stream>>>(xbuf, src, lengths, pbuf, wsbuf,
                                       out_a, cbuf);

  // out = tanh(concat(c, input) @ Wo^T)   (M=128, N=1024, K=2048)
  gemm_out_kernel<<<g1, 128, 0, stream>>>(cbuf, inp, Wo, out_o);
}